// TemporalPrototypeManager_52948356825776
// MI455X (gfx1250) — compile-verified
//
#include <hip/hip_runtime.h>
#include <stdint.h>

// ---------------------------------------------------------------------------
// TemporalPrototypeManager for MI455X (gfx1250)
//
// Pass 1 (bandwidth-bound, ~512MB of feature reads @ 23.3TB/s => ~22us floor):
//   grid = (128 row-chunks, 4 column-tiles of 32). Each block keeps a private
//   LDS histogram [1000 classes x 32 cols] (125KB; CDNA5 has 320KB/WGP so two
//   blocks co-reside). Features stream in via async global->LDS copies
//   (double buffered), then ds_add_f32 scatters them into the histogram with
//   zero bank conflicts (lane == column). Partials flush with
//   global_atomic_add_f32 into workspace sums/counts.
// Pass 2 (tiny): per 16 classes, one wave computes prototype row-sums with
//   V_WMMA_F32_16X16X4_F32 (P x ones row-reduction), derives per-class
//   alpha/beta, and writes out = alpha*proto + beta*sums.
// ---------------------------------------------------------------------------

#define N_ROWS   1048576
#define DDIM     128
#define CCLS     1000
#define TILE     32
#define NTILES   4            // DDIM / TILE
#define BLOCKS_X 128
#define THREADS  256          // 8 wave32
#define WAVES    8
#define ROWS_PER_BLOCK (N_ROWS / BLOCKS_X)     // 8192
#define ROWS_PER_WAVE  (ROWS_PER_BLOCK / WAVES) // 1024
#define BATCH    8
#define NBATCH   (ROWS_PER_WAVE / BATCH)       // 128

// LDS layout (float indices)
#define HIST_F        0
#define HIST_FLOATS   (CCLS * TILE)            // 32000
#define CNT_F         HIST_FLOATS              // 32000
#define CNT_FLOATS    CCLS                     // 1000
#define STAGE_F       33024                    // byte 132096, 128B aligned
#define STAGE_SLOT_F  (BATCH * TILE)           // 256 floats / slot
#define STAGE_WAVE_F  (2 * STAGE_SLOT_F)       // double buffered
#define SMEM_FLOATS   (STAGE_F + WAVES * STAGE_WAVE_F)  // 37120 -> 148480 B

static_assert(N_ROWS % (BLOCKS_X * WAVES * BATCH) == 0, "divisibility");

typedef __attribute__((ext_vector_type(2))) float v2f;
typedef __attribute__((ext_vector_type(8))) float v8f;

__global__ void zero_ws_kernel(float* __restrict__ ws, int n) {
  int i = blockIdx.x * blockDim.x + threadIdx.x;
  if (i < n) ws[i] = 0.0f;
}

__global__ __launch_bounds__(THREADS) void proto_accum_kernel(
    const float* __restrict__ features, const int* __restrict__ labels,
    float* __restrict__ sums, float* __restrict__ counts) {
  __shared__ float smemf[SMEM_FLOATS];

  const int tid  = threadIdx.x;
  const int lane = tid & 31;
  const int wave = tid >> 5;
  const int tile = blockIdx.y;          // column tile 0..3
  const bool tile0 = (tile == 0);

  // zero private histogram + counts
  for (int i = tid; i < HIST_FLOATS + CNT_FLOATS; i += THREADS) smemf[i] = 0.0f;
  __syncthreads();

  const int rowBase = blockIdx.x * ROWS_PER_BLOCK + wave * ROWS_PER_WAVE;

  // per-lane global byte address of (rowBase, tile*32 + lane)
  uint64_t gbase = (uint64_t)(uintptr_t)features +
                   ((uint64_t)rowBase * DDIM + (uint64_t)(tile * TILE + lane)) * 4ull;
  // flat-pointer low 32 bits == LDS byte offset (LDS aperture low bits are 0)
  const uint32_t stage_off0    = (uint32_t)(uintptr_t)&smemf[STAGE_F + wave * STAGE_WAVE_F];
  const uint32_t hist_lane_off = (uint32_t)(uintptr_t)&smemf[HIST_F] + (uint32_t)lane * 4u;
  const uint32_t cnt_off       = (uint32_t)(uintptr_t)&smemf[CNT_F];

  auto issue = [&](int b) {
    uint64_t ga = gbase + (uint64_t)b * (uint64_t)(BATCH * DDIM * 4);
    uint32_t lo = stage_off0 + ((b & 1) ? (uint32_t)(STAGE_SLOT_F * 4) : 0u);
#pragma unroll
    for (int r = 0; r < BATCH; ++r) {
      uint32_t l = lo + (uint32_t)(r * TILE * 4);
      uint64_t g = ga + (uint64_t)(r * DDIM * 4);
      asm volatile("global_load_async_to_lds_b32 %0, %1, off"
                   :: "v"(l), "v"(g) : "memory");
    }
  };

  issue(0);
  for (int b = 0; b < NBATCH; ++b) {
    // 8 labels for this batch (lanes 0..7 hold distinct rows)
    int lbl8 = labels[rowBase + b * BATCH + (lane & 7)];

    if (b + 1 < NBATCH) {
      issue(b + 1);
      asm volatile("s_wait_asynccnt 0x8" ::: "memory"); // batch b landed in LDS
    } else {
      asm volatile("s_wait_asynccnt 0x0" ::: "memory");
    }

    const int sbase = STAGE_F + wave * STAGE_WAVE_F + ((b & 1) ? STAGE_SLOT_F : 0);
#pragma unroll
    for (int r = 0; r < BATCH; ++r) {
      int   lbl = __builtin_amdgcn_readlane(lbl8, r);   // scalar label
      float v   = smemf[sbase + r * TILE + lane];       // ds_load_b32
      uint32_t haddr = hist_lane_off + (uint32_t)lbl * (uint32_t)(TILE * 4);
      asm volatile("ds_add_f32 %0, %1" :: "v"(haddr), "v"(v) : "memory");
    }
    if (tile0 && lane < 8) {                            // counts: +1 per row
      uint32_t caddr = cnt_off + (uint32_t)lbl8 * 4u;
      float one = 1.0f;
      asm volatile("ds_add_f32 %0, %1" :: "v"(caddr), "v"(one) : "memory");
    }
  }

  // inline-asm DS atomics are invisible to the compiler's waitcnt pass
  asm volatile("s_wait_dscnt 0x0" ::: "memory");
  __syncthreads();

  // flush private histogram into global partial sums (hits L2: 516KB working set)
  for (int idx = tid; idx < HIST_FLOATS; idx += THREADS) {
    float v  = smemf[HIST_F + idx];
    int c    = idx >> 5;
    int col  = idx & (TILE - 1);
    __hip_atomic_fetch_add(&sums[c * DDIM + tile * TILE + col], v,
                           __ATOMIC_RELAXED, __HIP_MEMORY_SCOPE_AGENT);
  }
  if (tile0) {
    for (int idx = tid; idx < CCLS; idx += THREADS) {
      __hip_atomic_fetch_add(&counts[idx], smemf[CNT_F + idx],
                             __ATOMIC_RELAXED, __HIP_MEMORY_SCOPE_AGENT);
    }
  }
}

// One wave handles 16 classes. Row-sum of prototypes computed as P x ones via
// V_WMMA_F32_16X16X4_F32 accumulated over K = 128. OOB rows (only the 8 padded
// classes of the last block) are CLAMPED instead of predicated so the A-matrix
// loads stay unconditional (no exec-mask churn, mergeable b64 loads); their
// garbage row-sums are never consumed because all uses are guarded by c < CCLS.
__global__ __launch_bounds__(32) void proto_finalize_kernel(
    const float* __restrict__ proto, const float* __restrict__ sums,
    const float* __restrict__ counts, float* __restrict__ out) {
  const int lane  = threadIdx.x;          // 0..31
  const int cbase = blockIdx.x * 16;
  const int m     = lane & 15;
  const int kh    = (lane >> 4) << 1;     // lanes 0-15 -> K {0,1}; 16-31 -> K {2,3}
  const int crow  = cbase + m;
  const int crow_c = (crow < CCLS) ? crow : (CCLS - 1);   // clamp, don't branch
  const float* arow = proto + (size_t)crow_c * DDIM + kh; // 8B aligned

  v8f acc = {0.f, 0.f, 0.f, 0.f, 0.f, 0.f, 0.f, 0.f};
  v2f bmat; bmat[0] = 1.0f; bmat[1] = 1.0f;   // ones matrix: layout-independent
#pragma unroll
  for (int k0 = 0; k0 < DDIM; k0 += 4) {
    v2f a = *(const v2f*)(arow + k0);         // unconditional global_load_b64
    acc = __builtin_amdgcn_wmma_f32_16x16x4_f32(false, a, false, bmat,
                                                (short)0, acc, false, false);
  }
  // D[m][n] == rowsum(m) for every n. VGPR r: lanes 0-15 -> M=r, lanes 16-31 -> M=8+r.
  __shared__ float s_rowsum[16];
  __shared__ float s_alpha[16];
  __shared__ float s_beta[16];
  if (lane == 0) {
#pragma unroll
    for (int r = 0; r < 8; ++r) s_rowsum[r] = acc[r];
  }
  if (lane == 16) {
#pragma unroll
    for (int r = 0; r < 8; ++r) s_rowsum[8 + r] = acc[r];
  }
  __syncthreads();
  if (lane < 16) {
    int c = cbase + lane;
    float cnt = (c < CCLS) ? counts[c] : 0.0f;
    float alpha, beta;
    if (!(cnt > 0.0f))              { alpha = 1.0f; beta = 0.0f; }        // no samples
    else if (s_rowsum[lane] == 0.f) { alpha = 0.0f; beta = 1.0f / cnt; }  // cold start
    else                            { alpha = 0.9f; beta = 0.1f / cnt; }  // momentum
    s_alpha[lane] = alpha;
    s_beta[lane]  = beta;
  }
  __syncthreads();
#pragma unroll 4
  for (int i = 0; i < (16 * DDIM) / 32; ++i) {
    int idx = i * 32 + lane;
    int ml  = idx >> 7;            // class-in-tile (uniform per i)
    int col = idx & (DDIM - 1);
    int c   = cbase + ml;
    if (c < CCLS) {
      size_t o = (size_t)c * DDIM + col;
      out[o] = s_alpha[ml] * proto[o] + s_beta[ml] * sums[o];
    }
  }
}

extern "C" void kernel_launch(void* const* d_in, const int* in_sizes, int n_in,
                              void* d_out, int out_size, void* d_ws, size_t ws_size,
                              hipStream_t stream) {
  const float* features = (const float*)d_in[0];
  const int*   labels   = (const int*)d_in[1];
  const float* proto    = (const float*)d_in[2];
  float* sums   = (float*)d_ws;                 // [C, D]
  float* counts = sums + (size_t)CCLS * DDIM;   // [C]
  float* out    = (float*)d_out;

  const int zn = CCLS * DDIM + CCLS;
  zero_ws_kernel<<<(zn + 255) / 256, 256, 0, stream>>>(sums, zn);

  proto_accum_kernel<<<dim3(BLOCKS_X, NTILES), THREADS, 0, stream>>>(
      features, labels, sums, counts);

  proto_finalize_kernel<<<(CCLS + 15) / 16, 32, 0, stream>>>(
      proto, sums, counts, out);
}